// WavetableSynthesizer_12137577578987
// MI455X (gfx1250) — compile-verified
//
#include <hip/hip_runtime.h>
#include <hip/hip_bf16.h>
#include <math.h>

// ---------------- problem constants ----------------
#define NB     16
#define NF     400
#define NTAB   20
#define LTAB   512
#define NBIN   257          // LTAB/2 + 1
#define BLOCK  240
#define TSAMP  (NF * BLOCK) // 96000
#define FSR    24000.0f
#define TWO_PI_F 6.28318530717958647692f
#define KMAXM  152          // bins kept (fc <= 150, pad to 152 -> K = 304)
#define KDIM   (2 * KMAXM)  // 304, multiple of 4 for 16x16x4 WMMA

typedef __attribute__((ext_vector_type(2))) float v2f;
typedef __attribute__((ext_vector_type(8))) float v8f;
typedef int v4i __attribute__((vector_size(16)));   // <4 x i32> for async-LDS builtin
typedef __attribute__((address_space(1))) v4i* gv4i_p;
typedef __attribute__((address_space(3))) v4i* lv4i_p;

// Async global->LDS (CDNA5 GLOBAL_LOAD_ASYNC_TO_LDS_B128, ASYNCcnt-tracked),
// guarded so compile never regresses if the toolchain lacks the builtins.
#if defined(__has_builtin)
#if __has_builtin(__builtin_amdgcn_global_load_async_to_lds_b128) && \
    __has_builtin(__builtin_amdgcn_s_wait_asynccnt)
#define HAVE_ASYNC_LDS 1
#endif
#endif

// ---------------------------------------------------------------------------
// Kernel 1: direct rDFT of the wavetable bank. X[tab][m] = sum_x wt[x] e^{-i2pi m x/512}
// 20*257 outputs, 512 MACs each -> ~5 MFLOP, trivial.
// ---------------------------------------------------------------------------
__global__ __launch_bounds__(128) void k_rfft(const float* __restrict__ wt,
                                              float* __restrict__ X) {
    int idx = blockIdx.x * blockDim.x + threadIdx.x;
    if (idx >= NTAB * NBIN) return;
    int tab = idx / NBIN;
    int m   = idx - tab * NBIN;
    const float* w = wt + tab * LTAB;
    float re = 0.0f, im = 0.0f;
    for (int x = 0; x < LTAB; ++x) {
        int t = (m * x) & (LTAB - 1);                 // exact periodic reduction
        float ang = -TWO_PI_F * (float)t * (1.0f / LTAB);
        float s, c;
        __sincosf(ang, &s, &c);
        re = fmaf(w[x], c, re);
        im = fmaf(w[x], s, im);
    }
    X[2 * idx + 0] = re;
    X[2 * idx + 1] = im;
}

// ---------------------------------------------------------------------------
// Kernel 2: build transposed basis Bt (512 cols x 304 K).
// Row pair k=2m -> scale*cos(2pi m x/512), k=2m+1 -> -scale*sin(...)
// scale = 1/512 for m==0 else 2/512 (irfft normalization of masked rfft).
// ---------------------------------------------------------------------------
__global__ __launch_bounds__(256) void k_build_bt(float* __restrict__ Bt) {
    int idx = blockIdx.x * blockDim.x + threadIdx.x;
    if (idx >= LTAB * KDIM) return;
    int x = idx / KDIM;
    int k = idx - x * KDIM;
    int m = k >> 1;
    float scale = ((m == 0) ? 1.0f : 2.0f) * (1.0f / LTAB);
    int t = (m * x) & (LTAB - 1);
    float ang = TWO_PI_F * (float)t * (1.0f / LTAB);
    Bt[idx] = (k & 1) ? (-scale * __sinf(ang)) : (scale * __cosf(ang));
}

// ---------------------------------------------------------------------------
// Kernel 3: build A = masked amplitude-combined spectra. One block per (b,f) row.
// A[row][2m]   = mask(m<fc) * sum_tab amp[tab]*Xre[tab][m]
// A[row][2m+1] = mask(m<fc) * sum_tab amp[tab]*Xim[tab][m]
// ---------------------------------------------------------------------------
__global__ __launch_bounds__(160) void k_build_a(const float* __restrict__ f0,
                                                 const float* __restrict__ amps,
                                                 const float* __restrict__ X,
                                                 float* __restrict__ A) {
    int row = blockIdx.x;                    // b*NF + f
    __shared__ float amp_s[NTAB];
    if (threadIdx.x < NTAB) amp_s[threadIdx.x] = amps[row * NTAB + threadIdx.x];
    __syncthreads();

    int m = threadIdx.x;
    if (m >= KMAXM) return;
    float f  = f0[row];
    int   fc = (int)ceilf(FSR / (2.0f * f)); // cutoff bin (<=150)
    float re = 0.0f, im = 0.0f;
    if (m < fc) {
        #pragma unroll 4
        for (int t = 0; t < NTAB; ++t) {
            const float* xp = X + 2 * (t * NBIN + m);
            re = fmaf(amp_s[t], xp[0], re);
            im = fmaf(amp_s[t], xp[1], im);
        }
    }
    A[row * KDIM + 2 * m + 0] = re;
    A[row * KDIM + 2 * m + 1] = im;
}

// ---------------------------------------------------------------------------
// Kernel 4: per-batch frame-start phase prefix scan (400 steps, 16 lanes).
// P[b][f] = sum_{g<f} BLOCK * (2pi f0[b,g]/FS).  final_phase = P[b][NF] mod 2pi.
// ---------------------------------------------------------------------------
__global__ __launch_bounds__(32) void k_phase_scan(const float* __restrict__ f0,
                                                   float* __restrict__ P,
                                                   float* __restrict__ final_phase) {
    int b = threadIdx.x;
    if (b >= NB) return;
    float acc = 0.0f;
    for (int f = 0; f < NF; ++f) {
        P[b * NF + f] = acc;
        float d = TWO_PI_F * f0[b * NF + f] * (1.0f / FSR);
        acc = fmaf((float)BLOCK, d, acc);
    }
    final_phase[b] = fmodf(acc, TWO_PI_F);
}

// ---------------------------------------------------------------------------
// Kernel 5: the heavy step as a WMMA f32 GEMM.
//   CT(6400x512) = A(6400x304) x B(304x512)      (B stored transposed 512x304)
// One workgroup (8 waves) per 16-row tile; wave w owns a 16x64 column strip.
// A tile (16x304 = 19.5 KB) staged into LDS with async b128 copies
// (ASYNCcnt), then the K loop issues v_wmma_f32_16x16x4_f32 per 16x16 tile.
// ---------------------------------------------------------------------------
__global__ __launch_bounds__(256) void k_gemm_ct(const float* __restrict__ A,
                                                 const float* __restrict__ Bt,
                                                 float* __restrict__ CT) {
    const int rowTile = blockIdx.x;          // 0..399 -> rows rowTile*16..+15
    const int wave    = threadIdx.x >> 5;    // 0..7
    const int lane    = threadIdx.x & 31;
    const int lr      = lane & 15;           // row-within-tile / col-within-16
    const int hi      = lane >> 4;           // half-wave select (K pair / M half)
    const int colBase = wave * 64;

    __shared__ float As[16 * KDIM];          // 19456 bytes
    const float* Ag = A + (size_t)rowTile * 16 * KDIM;

#if defined(HAVE_ASYNC_LDS)
    {
        // Each lane copies 16B global->LDS directly (no VGPR round trip);
        // one wave moves 512B per instruction; 19456B total per workgroup.
        char* gsrc = (char*)Ag;
        char* ldst = (char*)As;
        for (int off = (int)threadIdx.x * 16; off < 16 * KDIM * 4; off += 256 * 16) {
            __builtin_amdgcn_global_load_async_to_lds_b128(
                (gv4i_p)(gsrc + off), (lv4i_p)(ldst + off), 0, 0);
        }
        __builtin_amdgcn_s_wait_asynccnt(0);
    }
    __syncthreads();
#else
    for (int i = threadIdx.x; i < 16 * KDIM; i += 256) As[i] = Ag[i];
    __syncthreads();
#endif

    v8f acc0 = {}, acc1 = {}, acc2 = {}, acc3 = {};

    const float* b0 = Bt + (size_t)(colBase + 0 * 16 + lr) * KDIM;
    const float* b1 = Bt + (size_t)(colBase + 1 * 16 + lr) * KDIM;
    const float* b2 = Bt + (size_t)(colBase + 2 * 16 + lr) * KDIM;
    const float* b3 = Bt + (size_t)(colBase + 3 * 16 + lr) * KDIM;
    const float* ar = As + lr * KDIM;

    for (int k = 0; k < KDIM; k += 4) {
        int kk = k + 2 * hi;                 // ISA 16x4 f32 layout: VGPR0 K=0/K=2, VGPR1 K=1/K=3
        v2f a, f0v, f1v, f2v, f3v;
        a.x = ar[kk];     a.y = ar[kk + 1];
        f0v.x = b0[kk];   f0v.y = b0[kk + 1];
        f1v.x = b1[kk];   f1v.y = b1[kk + 1];
        f2v.x = b2[kk];   f2v.y = b2[kk + 1];
        f3v.x = b3[kk];   f3v.y = b3[kk + 1];
        acc0 = __builtin_amdgcn_wmma_f32_16x16x4_f32(false, a, false, f0v, (short)0, acc0, false, false);
        acc1 = __builtin_amdgcn_wmma_f32_16x16x4_f32(false, a, false, f1v, (short)0, acc1, false, false);
        acc2 = __builtin_amdgcn_wmma_f32_16x16x4_f32(false, a, false, f2v, (short)0, acc2, false, false);
        acc3 = __builtin_amdgcn_wmma_f32_16x16x4_f32(false, a, false, f3v, (short)0, acc3, false, false);
    }

    // C/D 16x16 layout: VGPR r -> M = r + 8*hi, N = lr (per half-wave)
    float* C = CT + (size_t)(rowTile * 16) * LTAB;
    #pragma unroll
    for (int r = 0; r < 8; ++r) {
        int mrow = r + 8 * hi;
        C[mrow * LTAB + colBase + 0 * 16 + lr] = acc0[r];
        C[mrow * LTAB + colBase + 1 * 16 + lr] = acc1[r];
        C[mrow * LTAB + colBase + 2 * 16 + lr] = acc2[r];
        C[mrow * LTAB + colBase + 3 * 16 + lr] = acc3[r];
    }
}

// ---------------------------------------------------------------------------
// Kernel 6: synthesis — phase -> position -> linear-interp lookup in the
// per-frame combined table CT[row], one thread per output sample.
// ---------------------------------------------------------------------------
__global__ __launch_bounds__(256) void k_synth(const float* __restrict__ f0,
                                               const float* __restrict__ P,
                                               const float* __restrict__ CT,
                                               float* __restrict__ out) {
    int t = blockIdx.x * blockDim.x + threadIdx.x;
    if (t >= NB * TSAMP) return;
    int b = t / TSAMP;
    int r = t - b * TSAMP;
    int f = r / BLOCK;
    int j = r - f * BLOCK;
    int row = b * NF + f;

    float d   = TWO_PI_F * f0[row] * (1.0f / FSR);
    float ph  = fmaf((float)(j + 1), d, P[row]);
    ph        = fmodf(ph, TWO_PI_F);
    float pos = ph * ((float)LTAB / TWO_PI_F);
    float bse = floorf(pos);
    float frc = pos - bse;
    int i0 = ((int)bse) & (LTAB - 1);
    int i1 = (i0 + 1) & (LTAB - 1);
    const float* ct = CT + (size_t)row * LTAB;
    float w0 = ct[i0];
    float w1 = ct[i1];
    out[t] = fmaf(frc, w1 - w0, fmaf(-frc, w0, w0));  // w0 + frc*(w1-2*w0)
}

// ---------------------------------------------------------------------------
extern "C" void kernel_launch(void* const* d_in, const int* in_sizes, int n_in,
                              void* d_out, int out_size, void* d_ws, size_t ws_size,
                              hipStream_t stream) {
    const float* f0   = (const float*)d_in[0];   // (16,400,1)
    const float* amps = (const float*)d_in[1];   // (16,400,20)
    const float* wt   = (const float*)d_in[2];   // (20,512)
    // d_in[3] (initial_phase) is unused by the reference computation.

    float* ws = (float*)d_ws;
    float* X  = ws;                              // 20*257*2      = 10,280
    float* Bt = X  + NTAB * NBIN * 2;            // 512*304       = 155,648
    float* A  = Bt + LTAB * KDIM;                // 6400*304      = 1,945,600
    float* CT = A  + (size_t)NB * NF * KDIM;     // 6400*512      = 3,276,800
    float* P  = CT + (size_t)NB * NF * LTAB;     // 6400
    // total ~ 21.6 MB of workspace

    float* sig         = (float*)d_out;          // (16, 96000)
    float* final_phase = sig + (size_t)NB * TSAMP; // (16,) appended

    k_rfft      <<<dim3((NTAB * NBIN + 127) / 128), dim3(128), 0, stream>>>(wt, X);
    k_build_bt  <<<dim3((LTAB * KDIM + 255) / 256), dim3(256), 0, stream>>>(Bt);
    k_build_a   <<<dim3(NB * NF), dim3(160), 0, stream>>>(f0, amps, X, A);
    k_phase_scan<<<dim3(1), dim3(32), 0, stream>>>(f0, P, final_phase);
    k_gemm_ct   <<<dim3(NB * NF / 16), dim3(256), 0, stream>>>(A, Bt, CT);
    k_synth     <<<dim3((NB * TSAMP + 255) / 256), dim3(256), 0, stream>>>(f0, P, CT, sig);
}